// GAT_59493886984410
// MI455X (gfx1250) — compile-verified
//
#include <hip/hip_runtime.h>
#include <hip/hip_bf16.h>
#include <math.h>

#define N_NODES   100000
#define N_EDGES   800000
#define IN_DIM    64
#define HEADS     8
#define OUT_C     64
#define HC        512          // HEADS * OUT_C
#define NEG_SLOPE 0.2f

typedef float v2f __attribute__((ext_vector_type(2)));
typedef float v8f __attribute__((ext_vector_type(8)));

// ---- order-preserving float<->uint mapping for atomic segment-max ----
__device__ __forceinline__ unsigned f32_to_ord(float f) {
    unsigned b = __float_as_uint(f);
    return (b & 0x80000000u) ? ~b : (b | 0x80000000u);
}
__device__ __forceinline__ float ord_to_f32(unsigned e) {
    unsigned b = (e & 0x80000000u) ? (e ^ 0x80000000u) : ~e;
    return __uint_as_float(b);
}
#define ORD_NEG_INF 0x007FFFFFu   // f32_to_ord(-inf)

// ============================================================================
// Kernel 1: xl = x @ W_l, xr = x @ W_r via V_WMMA_F32_16X16X4_F32.
// Block = 256 threads (8 waves), handles 16 node rows. 64 output 16x16 tiles
// (32 col-tiles for W_l + 32 for W_r); wave w owns tiles w, w+8, ..., w+56.
// A-slice (16 rows x 64 K) is loaded once per wave and reused for all tiles.
// ============================================================================
__global__ __launch_bounds__(256) void gat_gemm_wmma(
    const float* __restrict__ x, const float* __restrict__ Wl,
    const float* __restrict__ Wr, float* __restrict__ xl,
    float* __restrict__ xr)
{
    const int lane    = threadIdx.x & 31;
    const int wave    = threadIdx.x >> 5;
    const int rowbase = blockIdx.x * 16;

    // A layout (16x4 f32): lanes 0-15 -> M=lane, K={k0,k0+1}; lanes 16-31 -> K={k0+2,k0+3}
    const int arow = lane & 15;
    const int kk   = (lane >> 4) << 1;       // 0 or 2

    v2f a[16];
    const float* xrow = x + (size_t)(rowbase + arow) * IN_DIM;
    #pragma unroll
    for (int ks = 0; ks < 16; ++ks) {
        const int k = ks * 4 + kk;
        a[ks].x = xrow[k];
        a[ks].y = xrow[k + 1];
    }

    const int ncol = lane & 15;
    for (int t = wave; t < 64; t += 8) {
        const float* W   = (t < 32) ? Wl : Wr;
        float*       dst = (t < 32) ? xl : xr;
        const int colbase = (t & 31) * 16;
        const int col     = colbase + ncol;

        v8f acc = {0.f, 0.f, 0.f, 0.f, 0.f, 0.f, 0.f, 0.f};
        #pragma unroll
        for (int ks = 0; ks < 16; ++ks) {
            const int k = ks * 4 + kk;
            v2f b;                                   // B (4x16): row K striped over lanes
            b.x = W[(size_t)k       * HC + col];
            b.y = W[(size_t)(k + 1) * HC + col];
            acc = __builtin_amdgcn_wmma_f32_16x16x4_f32(
                      false, a[ks], false, b, (short)0, acc, false, false);
        }

        // D layout: VGPR v -> lanes 0-15: M=v, lanes 16-31: M=v+8; N = lane%16
        const int mbase = (lane >> 4) << 3;          // 0 or 8
        #pragma unroll
        for (int v = 0; v < 8; ++v) {
            const int m = mbase + v;
            dst[(size_t)(rowbase + m) * HC + col] = acc[v];
        }
    }
}

// ============================================================================
// Kernel 2: init out = bias (handles zero-in-edge nodes + folds bias),
// m_enc = enc(-inf), denom = 0. Runs every launch (graph-replay determinism).
// ============================================================================
__global__ void gat_init(float* __restrict__ out, const float* __restrict__ bias,
                         unsigned* __restrict__ m_enc, float* __restrict__ denom)
{
    const size_t tid = (size_t)blockIdx.x * blockDim.x + threadIdx.x;
    if (tid < (size_t)N_NODES * HC) out[tid] = bias[tid & (HC - 1)];
    if (tid < (size_t)N_NODES * HEADS) {
        m_enc[tid] = ORD_NEG_INF;
        denom[tid] = 0.f;
    }
}

// ============================================================================
// Kernel 3: per-edge GATv2 score e[E,H] = att_h . leakyrelu(xl[src]+xr[dst]),
// plus atomic segment-max into m_enc. One wave32 per edge; coalesced loads.
// ============================================================================
__global__ __launch_bounds__(256) void gat_edge_score(
    const long long* __restrict__ ei, const float* __restrict__ xl,
    const float* __restrict__ xr, const float* __restrict__ att,
    float* __restrict__ e_buf, unsigned* __restrict__ m_enc)
{
    const int edge = blockIdx.x * 8 + (threadIdx.x >> 5);
    if (edge >= N_EDGES) return;
    const int lane = threadIdx.x & 31;

    const long long src = ei[edge];
    const long long dst = ei[N_EDGES + edge];
    const float* pl = xl + (size_t)src * HC;
    const float* pr = xr + (size_t)dst * HC;

    #pragma unroll
    for (int h = 0; h < HEADS; ++h) {
        const int c0 = h * OUT_C + lane;
        float v0 = pl[c0] + pr[c0];
        v0 = (v0 > 0.f) ? v0 : NEG_SLOPE * v0;
        float p = v0 * att[c0];
        const int c1 = c0 + 32;
        float v1 = pl[c1] + pr[c1];
        v1 = (v1 > 0.f) ? v1 : NEG_SLOPE * v1;
        p += v1 * att[c1];
        #pragma unroll
        for (int off = 16; off > 0; off >>= 1) p += __shfl_xor(p, off, 32);
        if (lane == 0) {
            e_buf[(size_t)edge * HEADS + h] = p;
            atomicMax(&m_enc[(size_t)dst * HEADS + h], f32_to_ord(p));
        }
    }
}

// ============================================================================
// Kernel 4: decode m_enc -> float in place; -inf (no in-edges) -> 0.
// ============================================================================
__global__ void gat_finalize_m(unsigned* __restrict__ m)
{
    const int tid = blockIdx.x * blockDim.x + threadIdx.x;
    if (tid >= N_NODES * HEADS) return;
    float f = ord_to_f32(m[tid]);
    if (!isfinite(f)) f = 0.f;
    m[tid] = __float_as_uint(f);
}

// ============================================================================
// Kernel 5: ex = exp(e - m[dst]); e_buf <- ex; denom[dst] += ex (atomic).
// ============================================================================
__global__ void gat_edge_exp(const long long* __restrict__ ei,
                             float* __restrict__ e_buf,
                             const float* __restrict__ m_f,
                             float* __restrict__ denom)
{
    const int tid = blockIdx.x * blockDim.x + threadIdx.x;
    if (tid >= N_EDGES * HEADS) return;
    const int edge = tid >> 3;
    const int h    = tid & 7;
    const long long dst = ei[N_EDGES + edge];
    const float ex = __expf(e_buf[tid] - m_f[(size_t)dst * HEADS + h]);
    e_buf[tid] = ex;
    atomicAdd(&denom[(size_t)dst * HEADS + h], ex);
}

// ============================================================================
// Kernel 6: out[dst] += alpha * xl[src] (fp32 global atomics). Wave32/edge;
// lanes 0-7 compute per-head alpha, broadcast via shuffle.
// ============================================================================
__global__ __launch_bounds__(256) void gat_scatter(
    const long long* __restrict__ ei, const float* __restrict__ xl,
    const float* __restrict__ e_buf, const float* __restrict__ denom,
    float* __restrict__ out)
{
    const int edge = blockIdx.x * 8 + (threadIdx.x >> 5);
    if (edge >= N_EDGES) return;
    const int lane = threadIdx.x & 31;

    const long long src = ei[edge];
    const long long dst = ei[N_EDGES + edge];

    float alpha = 0.f;
    if (lane < HEADS)
        alpha = e_buf[(size_t)edge * HEADS + lane] /
                (denom[(size_t)dst * HEADS + lane] + 1e-16f);

    const float* pl = xl + (size_t)src * HC;
    float*       po = out + (size_t)dst * HC;

    #pragma unroll
    for (int h = 0; h < HEADS; ++h) {
        const float ah = __shfl(alpha, h, 32);
        const int c0 = h * OUT_C + lane;
        atomicAdd(&po[c0],      ah * pl[c0]);
        atomicAdd(&po[c0 + 32], ah * pl[c0 + 32]);
    }
}

// ============================================================================
extern "C" void kernel_launch(void* const* d_in, const int* in_sizes, int n_in,
                              void* d_out, int out_size, void* d_ws, size_t ws_size,
                              hipStream_t stream)
{
    const float*     x    = (const float*)d_in[0];
    const long long* ei   = (const long long*)d_in[1];   // int64 edge_index [2,E]
    const float*     Wl   = (const float*)d_in[2];
    const float*     Wr   = (const float*)d_in[3];
    const float*     att  = (const float*)d_in[4];
    const float*     bias = (const float*)d_in[5];
    float*           out  = (float*)d_out;

    // workspace layout (floats)
    float* ws    = (float*)d_ws;
    float* xl    = ws;                                   // N*HC
    float* xr    = xl + (size_t)N_NODES * HC;            // N*HC
    float* e_buf = xr + (size_t)N_NODES * HC;            // E*H
    float* m     = e_buf + (size_t)N_EDGES * HEADS;      // N*H (encoded, then float)
    float* denom = m + (size_t)N_NODES * HEADS;          // N*H

    gat_gemm_wmma<<<N_NODES / 16, 256, 0, stream>>>(x, Wl, Wr, xl, xr);
    gat_init<<<(N_NODES * (HC / 256)), 256, 0, stream>>>(out, bias, (unsigned*)m, denom);
    gat_edge_score<<<(N_EDGES + 7) / 8, 256, 0, stream>>>(ei, xl, xr, att, e_buf, (unsigned*)m);
    gat_finalize_m<<<(N_NODES * HEADS + 255) / 256, 256, 0, stream>>>((unsigned*)m);
    gat_edge_exp<<<(N_EDGES * HEADS + 255) / 256, 256, 0, stream>>>(ei, e_buf, m, denom);
    gat_scatter<<<(N_EDGES + 7) / 8, 256, 0, stream>>>(ei, xl, e_buf, denom, out);
}